// GNAE_ENC_4827543240747
// MI455X (gfx1250) — compile-verified
//
#include <hip/hip_runtime.h>
#include <math.h>

typedef __attribute__((ext_vector_type(2))) float v2f;
typedef __attribute__((ext_vector_type(8))) float v8f;

#define IN_DIM 256
#define HID 96
#define OUT_DIM 64
#define ALPHA_F 0.15f
#define OMA_F 0.85f
#define EPS_F 1e-12f

// ---------------- CSR build: count, scan, fill ----------------

__global__ void zero_int_k(int* p, int n) {
    int i = blockIdx.x * blockDim.x + threadIdx.x;
    if (i < n) p[i] = 0;
}

__global__ void cnt_k(const int* __restrict__ col, int* cnt, int E) {
    int i = blockIdx.x * blockDim.x + threadIdx.x;
    if (i < E) atomicAdd(&cnt[col[i]], 1);
}

__global__ void dinv_from_cnt_k(const int* __restrict__ cnt, float* dinv, int n) {
    int i = blockIdx.x * blockDim.x + threadIdx.x;
    if (i < n) dinv[i] = rsqrtf((float)(cnt[i] + 1));  // +1 self loop; deg >= 1
}

// single-workgroup exclusive scan: rowptr[i] = sum_{j<i} cnt[j]; rowptr[n] = total
__global__ __launch_bounds__(1024) void scan_k(const int* __restrict__ cnt,
                                               int* __restrict__ rowptr, int n) {
    __shared__ int buf[1024];
    __shared__ int carry_s;
    const int tid = threadIdx.x;
    if (tid == 0) carry_s = 0;
    __syncthreads();
    for (int base = 0; base < n; base += 1024) {
        const int idx = base + tid;
        const int v = (idx < n) ? cnt[idx] : 0;
        buf[tid] = v;
        __syncthreads();
        for (int off = 1; off < 1024; off <<= 1) {
            const int t = (tid >= off) ? buf[tid - off] : 0;
            __syncthreads();
            buf[tid] += t;
            __syncthreads();
        }
        const int incl  = buf[tid];
        const int carry = carry_s;
        if (idx < n) rowptr[idx] = carry + incl - v;
        __syncthreads();
        if (tid == 1023) carry_s = carry + buf[1023];
        __syncthreads();
    }
    if (tid == 0) rowptr[n] = carry_s;
}

__global__ void copy_int_k(int* __restrict__ dst, const int* __restrict__ src, int n) {
    int i = blockIdx.x * blockDim.x + threadIdx.x;
    if (i < n) dst[i] = src[i];
}

// srcs[] gets, for each dst node (CSR over col), the list of source rows
__global__ void fill_k(const int* __restrict__ row, const int* __restrict__ col,
                       int* cursor, int* __restrict__ srcs, int E) {
    int i = blockIdx.x * blockDim.x + threadIdx.x;
    if (i < E) {
        const int p = atomicAdd(&cursor[col[i]], 1);
        srcs[p] = row[i];
    }
}

// ---------------- dense GEMM via f32 WMMA ----------------
// Out[nrows x M] = X[nrows x K] @ W[K x M] + bias[M],  M = NT*16
// One wave computes a 16-row x M strip: A fragment reused across NT WMMAs.
// A frag (16x4 f32): lane l, reg r -> A[l&15][2*(l>>4)+r]
// B frag (4x16 f32): lane l, reg r -> B[2*(l>>4)+r][l&15]
// C/D (16x16 f32):   lane l, reg r -> C[r + 8*(l>>4)][l&15]
template <int NT>
__global__ __launch_bounds__(256) void gemm_wmma_k(
    const float* __restrict__ X, const float* __restrict__ W,
    const float* __restrict__ bias, float* __restrict__ Out,
    int K, int nrows) {
    const int M    = NT * 16;
    const int wave = threadIdx.x >> 5;
    const int tm   = blockIdx.x * 8 + wave;   // 16-row tile index
    if (tm * 16 >= nrows) return;
    const int lane = threadIdx.x & 31;
    const int half = lane >> 4;
    const int l16  = lane & 15;

    v8f c[NT];
#pragma unroll
    for (int t = 0; t < NT; ++t) c[t] = (v8f){};

    const float* __restrict__ xrow = X + (size_t)(tm * 16 + l16) * K;

    for (int k = 0; k < K; k += 4) {
        const int kk = k + 2 * half;
        v2f a;
        a.x = xrow[kk];
        a.y = xrow[kk + 1];
        const float* __restrict__ w0 = W + (size_t)kk * M + l16;
#pragma unroll
        for (int t = 0; t < NT; ++t) {
            v2f b;
            b.x = w0[t * 16];
            b.y = w0[M + t * 16];
            c[t] = __builtin_amdgcn_wmma_f32_16x16x4_f32(
                false, a, false, b, (short)0, c[t], false, false);
        }
    }

#pragma unroll
    for (int t = 0; t < NT; ++t) {
        const int bcol = t * 16 + l16;
        const float bv = bias[bcol];
#pragma unroll
        for (int r = 0; r < 8; ++r) {
            const int orow = tm * 16 + r + 8 * half;
            Out[(size_t)orow * M + bcol] = c[t][r] + bv;
        }
    }
}

// ---------------- row-wise L2 normalize * scale ----------------
__global__ __launch_bounds__(256) void l2norm_k(float* x, int n, int M, float s) {
    const int wave = threadIdx.x >> 5;
    const int lane = threadIdx.x & 31;
    const int row  = blockIdx.x * 8 + wave;
    if (row >= n) return;
    float* p = x + (size_t)row * M;
    float sum = 0.0f;
    for (int f = lane; f < M; f += 32) { float v = p[f]; sum += v * v; }
#pragma unroll
    for (int o = 16; o > 0; o >>= 1) sum += __shfl_xor(sum, o, 32);
    const float scale = s / fmaxf(sqrtf(sum), EPS_F);
    for (int f = lane; f < M; f += 32) p[f] *= scale;
}

// ---------------- APPNP step: prescale + CSR gather (no atomics) ----------------

// Y[i][f] = x[i][f] * dinv[i]
__global__ void prescale_k(float* __restrict__ Y, const float* __restrict__ x,
                           const float* __restrict__ dinv, int n, int M) {
    size_t i = (size_t)blockIdx.x * blockDim.x + threadIdx.x;
    if (i >= (size_t)n * M) return;
    Y[i] = x[i] * dinv[(int)(i / M)];
}

// dst[node] = 0.85 * dinv[node] * (Y[node] + sum_{src in CSR(node)} Y[src]) + 0.15 * h[node]
__global__ __launch_bounds__(256) void gather_prop_k(
    float* __restrict__ dst, const float* __restrict__ Y,
    const float* __restrict__ h, const float* __restrict__ dinv,
    const int* __restrict__ rowptr, const int* __restrict__ srcs,
    int n, int M, int relu) {
    const int wave = threadIdx.x >> 5;
    const int lane = threadIdx.x & 31;
    const int node = blockIdx.x * 8 + wave;
    if (node >= n) return;

    const float* __restrict__ yn = Y + (size_t)node * M;
    float a0 = yn[lane];
    float a1 = (M > 32) ? yn[lane + 32] : 0.0f;
    float a2 = (M > 64) ? yn[lane + 64] : 0.0f;

    const int e0 = rowptr[node];
    const int e1 = rowptr[node + 1];
    for (int e = e0; e < e1; ++e) {
        const float* __restrict__ ys = Y + (size_t)srcs[e] * M;
        a0 += ys[lane];
        if (M > 32) a1 += ys[lane + 32];
        if (M > 64) a2 += ys[lane + 64];
    }

    const float dd = OMA_F * dinv[node];
    const float* __restrict__ hn = h + (size_t)node * M;
    float* __restrict__ dn = dst + (size_t)node * M;

    float v0 = dd * a0 + ALPHA_F * hn[lane];
    if (relu) v0 = fmaxf(v0, 0.0f);
    dn[lane] = v0;
    if (M > 32) {
        float v1 = dd * a1 + ALPHA_F * hn[lane + 32];
        if (relu) v1 = fmaxf(v1, 0.0f);
        dn[lane + 32] = v1;
    }
    if (M > 64) {
        float v2 = dd * a2 + ALPHA_F * hn[lane + 64];
        if (relu) v2 = fmaxf(v2, 0.0f);
        dn[lane + 64] = v2;
    }
}

// ---------------- driver ----------------

static inline size_t align256(size_t b) { return (b + 255) & ~(size_t)255; }

extern "C" void kernel_launch(void* const* d_in, const int* in_sizes, int n_in,
                              void* d_out, int out_size, void* d_ws, size_t ws_size,
                              hipStream_t stream) {
    const float* x_in = (const float*)d_in[0];
    const int*   ei   = (const int*)d_in[1];     // [2 x E]: row then col
    const float* W0   = (const float*)d_in[2];
    const float* b0   = (const float*)d_in[3];
    const float* W1   = (const float*)d_in[4];
    const float* b1   = (const float*)d_in[5];
    const float* Wx   = (const float*)d_in[6];
    const float* bx   = (const float*)d_in[7];
    float* out = (float*)d_out;

    const int n = in_sizes[0] / IN_DIM;   // 50000
    const int E = in_sizes[1] / 2;        // 800000
    const int* row = ei;
    const int* col = ei + E;

    // workspace layout
    char* ws = (char*)d_ws;
    size_t off = 0;
    float* dinv   = (float*)(ws + off); off += align256((size_t)n * 4);
    int*   cnt    = (int*)  (ws + off); off += align256((size_t)n * 4);
    int*   rowptr = (int*)  (ws + off); off += align256((size_t)(n + 1) * 4);
    int*   cursor = (int*)  (ws + off); off += align256((size_t)n * 4);
    int*   srcs   = (int*)  (ws + off); off += align256((size_t)E * 4);
    float* H      = (float*)(ws + off); off += align256((size_t)n * HID * 4);
    float* X      = (float*)(ws + off); off += align256((size_t)n * HID * 4);
    float* Y      = (float*)(ws + off); off += align256((size_t)n * HID * 4);
    (void)ws_size;

    const int TPB  = 256;
    const int gN   = (n + TPB - 1) / TPB;
    const int gE   = (E + TPB - 1) / TPB;
    const int gRow = (n + 7) / 8;           // 8 waves/block, wave-per-row/node
    const int gTile = ((n / 16) + 7) / 8;   // 8 row-tiles per block in gemm

    // ---- CSR build + dinv ----
    zero_int_k<<<gN, TPB, 0, stream>>>(cnt, n);
    cnt_k<<<gE, TPB, 0, stream>>>(col, cnt, E);
    dinv_from_cnt_k<<<gN, TPB, 0, stream>>>(cnt, dinv, n);
    scan_k<<<1, 1024, 0, stream>>>(cnt, rowptr, n);
    copy_int_k<<<gN, TPB, 0, stream>>>(cursor, rowptr, n);
    fill_k<<<gE, TPB, 0, stream>>>(row, col, cursor, srcs, E);

    // one APPNP step: src -> dst with residual h (all row-major n x M)
    auto prop = [&](const float* src, float* dst, const float* h, int M, int relu) {
        const size_t total = (size_t)n * M;
        const int gT = (int)((total + TPB - 1) / TPB);
        prescale_k<<<gT, TPB, 0, stream>>>(Y, src, dinv, n, M);
        gather_prop_k<<<gRow, TPB, 0, stream>>>(dst, Y, h, dinv, rowptr, srcs, n, M, relu);
    };

    // ---- Layer 0: X_in(256) -> H(96), l2norm*1.8, APPNP x2, ReLU ----
    gemm_wmma_k<HID / 16><<<gTile, TPB, 0, stream>>>(x_in, W0, b0, H, IN_DIM, n);
    l2norm_k<<<gRow, TPB, 0, stream>>>(H, n, HID, 1.8f);
    prop(H, X, H, HID, 0);
    prop(X, X, H, HID, 1);   // relu(conv0 output)

    // ---- Layer 1: X(96) -> H(96), l2norm*1.8, APPNP x2, l2norm*1.5 ----
    gemm_wmma_k<HID / 16><<<gTile, TPB, 0, stream>>>(X, W1, b1, H, HID, n);
    l2norm_k<<<gRow, TPB, 0, stream>>>(H, n, HID, 1.8f);
    prop(H, X, H, HID, 0);
    prop(X, X, H, HID, 0);
    l2norm_k<<<gRow, TPB, 0, stream>>>(X, n, HID, 1.5f);

    // ---- Layer X: X(96) -> H(64), l2norm*1.8, APPNP x2 -> d_out ----
    gemm_wmma_k<OUT_DIM / 16><<<gTile, TPB, 0, stream>>>(X, Wx, bx, H, HID, n);
    l2norm_k<<<gRow, TPB, 0, stream>>>(H, n, OUT_DIM, 1.8f);
    prop(H, X, H, OUT_DIM, 0);
    prop(X, out, H, OUT_DIM, 0);
}